// LayerSummarizeNet_15264313770592
// MI455X (gfx1250) — compile-verified
//
#include <hip/hip_runtime.h>

// LSTM scan, MI455X (gfx1250, wave32).
// Wave 1 (producer): input chunks staged via GLOBAL_LOAD_ASYNC_TO_LDS (ASYNCcnt),
//   input projection via V_WMMA_F32_16X16X4_F32 (K = 3 inputs + 1 bias lane),
//   x_gates double-buffered into LDS.
// Wave 0 (consumer): serial recurrence; lane j owns gate element j; h broadcast
//   via v_readlane; split accumulator chains to cut dependence depth.

typedef __attribute__((ext_vector_type(2))) float v2f;
typedef __attribute__((ext_vector_type(8))) float v8f;
typedef int i32x4 __attribute__((vector_size(4 * sizeof(int))));
typedef int i32x2 __attribute__((vector_size(2 * sizeof(int))));

#define CHUNK 64   // timesteps per pipeline chunk
#define HID   20
#define NGATE 80   // 4*HID
#define NOUT  16
#define NSB   4    // input staging ring depth (chunks)
#define SROW  (CHUNK * 3 + 4)   // padded staging row (floats)

#if __has_builtin(__builtin_amdgcn_global_load_async_to_lds_b128) && \
    __has_builtin(__builtin_amdgcn_global_load_async_to_lds_b64)
#define HAVE_ASYNC_LDS 1
#else
#define HAVE_ASYNC_LDS 0
#endif

__device__ __forceinline__ float lane_bcast(float v, int l) {
  return __builtin_bit_cast(float, __builtin_amdgcn_readlane(__builtin_bit_cast(int, v), l));
}

__device__ __forceinline__ float fast_exp2(float x) {
#if __has_builtin(__builtin_amdgcn_exp2f)
  return __builtin_amdgcn_exp2f(x);   // v_exp_f32
#else
  return exp2f(x);
#endif
}

__device__ __forceinline__ float fast_sigmoid(float x) {
  // 1 / (1 + 2^(-x*log2 e))  -> v_exp_f32 + v_rcp_f32
  return __builtin_amdgcn_rcpf(1.0f + fast_exp2(-1.4426950408889634f * x));
}

__device__ __forceinline__ float fast_tanh(float x) {
#if __has_builtin(__builtin_amdgcn_tanhf)
  return __builtin_amdgcn_tanhf(x);   // v_tanh_f32 (gfx1250 trans op)
#else
  return __builtin_amdgcn_rcpf(1.0f + fast_exp2(-2.8853900817779268f * x)) * 2.0f - 1.0f;
#endif
}

__global__ __launch_bounds__(64, 1)
void lstm_scan_kernel(const float* __restrict__ input,   // [T,3]
                      const float* __restrict__ W_ih,    // [80,3]
                      const float* __restrict__ W_hh,    // [80,20]
                      const float* __restrict__ b_ih,    // [80]
                      const float* __restrict__ b_hh,    // [80]
                      const float* __restrict__ W_lin,   // [16,20]
                      const float* __restrict__ b_lin,   // [16]
                      float* __restrict__ out,           // [16]
                      int T) {
  __shared__ float xg[2][CHUNK][NGATE];       // 40 KB x_gates double buffer
#if HAVE_ASYNC_LDS
  __shared__ float instage[NSB][SROW];        // ~3 KB raw-input staging ring
#endif

  const int tid  = threadIdx.x;
  const int wave = tid >> 5;
  const int lane = tid & 31;
  const int nchunks = T / CHUNK;              // T = 1M, divisible by 64

  if (wave == 1) {
    // ------------------- producer: WMMA input projection -------------------
    // B tile (4x16 f32, 2 VGPRs/lane): VGPR0 = {K=0 | K=2}, VGPR1 = {K=1 | K=3}.
    // Row K=3 of B carries (b_ih + b_hh); A column K=3 is constant 1.0.
    v2f bt[5];
    const int col = lane & 15;
    const int kk  = (lane < 16) ? 0 : 2;
#pragma unroll
    for (int nt = 0; nt < 5; ++nt) {
      const int g = nt * 16 + col;
      bt[nt].x = W_ih[g * 3 + kk];
      bt[nt].y = (kk == 0) ? W_ih[g * 3 + 1] : (b_ih[g] + b_hh[g]);
    }

#if HAVE_ASYNC_LDS
    // Stage chunk c's 768 B of input into the ring: full-wave B128 (512 B)
    // + full-wave B64 (256 B); tracked by ASYNCcnt (2 instructions/chunk).
    auto stage = [&](int c) {
      const float* g = input + c * (CHUNK * 3);
      float* s = &instage[c & (NSB - 1)][0];
      __builtin_amdgcn_global_load_async_to_lds_b128(
          (i32x4*)(g + 4 * lane), (i32x4*)(s + 4 * lane), 0, 0);
      __builtin_amdgcn_global_load_async_to_lds_b64(
          (i32x2*)(g + 128 + 2 * lane), (i32x2*)(s + 128 + 2 * lane), 0, 0);
    };
#endif

    auto produce = [&](int c, int buf) {
#if HAVE_ASYNC_LDS
      // Allow the newest stage (2 asyncs) to remain in flight; chunk c is done.
      asm volatile("s_wait_asynccnt 0x2" ::: "memory");
      const float* st = &instage[c & (NSB - 1)][0];
#else
      const int t0 = c * CHUNK;
      if (t0 + 2 * CHUNK <= T)
        __builtin_prefetch(input + (t0 + CHUNK) * 3, 0, 3);
#endif
#pragma unroll
      for (int m = 0; m < CHUNK / 16; ++m) {
        // A tile 16x4 f32: VGPR0 = {K=0 | K=2}, VGPR1 = {K=1 | K=3}.
        v2f a;
#if HAVE_ASYNC_LDS
        const int base = (m * 16 + col) * 3 + kk;       // padded row: base+1 safe
        const float ax  = st[base];
        const float ay1 = st[base + 1];
        a.x = ax;
        a.y = (lane < 16) ? ay1 : 1.0f;                 // v_cndmask, no branch
#else
        const int t = c * CHUNK + m * 16 + col;
        if (lane < 16) { a.x = input[t * 3 + 0]; a.y = input[t * 3 + 1]; }
        else           { a.x = input[t * 3 + 2]; a.y = 1.0f; }
#endif
        const int rbase = m * 16 + ((lane >> 4) << 3);
#pragma unroll
        for (int nt = 0; nt < 5; ++nt) {
          v8f acc = {};
          v8f d = __builtin_amdgcn_wmma_f32_16x16x4_f32(
              false, a, false, bt[nt], (short)0, acc, false, false);
          const int cg = nt * 16 + col;
#pragma unroll
          for (int r = 0; r < 8; ++r)          // D: VGPR r = rows {r | 8+r}
            xg[buf][rbase + r][cg] = d[r];
        }
      }
#if HAVE_ASYNC_LDS
      if (c + 2 < nchunks) stage(c + 2);       // refill ring two chunks ahead
#endif
    };

#if HAVE_ASYNC_LDS
    stage(0);
    if (nchunks > 1) stage(1);
#endif
    produce(0, 0);
    __syncthreads();
    for (int c = 0; c < nchunks; ++c) {
      if (c + 1 < nchunks) produce(c + 1, (c + 1) & 1);
      __syncthreads();
    }
  } else {
    // ------------------------ consumer: serial scan -------------------------
    const int jj = (lane < HID) ? lane : 0;    // lanes 20..31 shadow lane 0
    float wi[HID], wf[HID], wg[HID], wo[HID];
#pragma unroll
    for (int k = 0; k < HID; ++k) {
      wi[k] = W_hh[(0 * HID + jj) * HID + k];
      wf[k] = W_hh[(1 * HID + jj) * HID + k];
      wg[k] = W_hh[(2 * HID + jj) * HID + k];
      wo[k] = W_hh[(3 * HID + jj) * HID + k];
    }
    float hv = 0.0f, cv = 0.0f;

    __syncthreads();   // matches producer prologue barrier
    for (int c = 0; c < nchunks; ++c) {
      const float* xb = &xg[c & 1][0][0];
      for (int s = 0; s < CHUNK; ++s) {
        const float* row = xb + s * NGATE;
        // Issue x_gate loads first; they join the chains only at the end,
        // so DS latency hides under the readlane/FMA stream.
        const float xi = row[0 * HID + jj];
        const float xf = row[1 * HID + jj];
        const float xq = row[2 * HID + jj];
        const float xo = row[3 * HID + jj];
        // Two 10-deep partial chains per gate (manual reassociation).
        float a0 = 0.f, a1 = 0.f, f0 = 0.f, f1 = 0.f;
        float g0 = 0.f, g1 = 0.f, o0 = 0.f, o1 = 0.f;
#pragma unroll
        for (int k = 0; k < HID; k += 2) {     // h broadcast: v_readlane -> SGPR
          const float h0 = lane_bcast(hv, k);
          const float h1 = lane_bcast(hv, k + 1);
          a0 = fmaf(h0, wi[k], a0);  a1 = fmaf(h1, wi[k + 1], a1);
          f0 = fmaf(h0, wf[k], f0);  f1 = fmaf(h1, wf[k + 1], f1);
          g0 = fmaf(h0, wg[k], g0);  g1 = fmaf(h1, wg[k + 1], g1);
          o0 = fmaf(h0, wo[k], o0);  o1 = fmaf(h1, wo[k + 1], o1);
        }
        const float ig = fast_sigmoid((a0 + a1) + xi);
        const float fg = fast_sigmoid((f0 + f1) + xf);
        const float gg = fast_tanh((g0 + g1) + xq);
        const float og = fast_sigmoid((o0 + o1) + xo);
        cv = fmaf(fg, cv, ig * gg);
        hv = og * fast_tanh(cv);
      }
      __syncthreads();
    }

    // final linear + sigmoid on h_T (lanes 0..15)
    if (lane < NOUT) {
      float acc = b_lin[lane];
#pragma unroll
      for (int k = 0; k < HID; ++k)
        acc = fmaf(lane_bcast(hv, k), W_lin[lane * HID + k], acc);
      out[lane] = fast_sigmoid(acc);
    }
  }
}

extern "C" void kernel_launch(void* const* d_in, const int* in_sizes, int n_in,
                              void* d_out, int out_size, void* d_ws, size_t ws_size,
                              hipStream_t stream) {
  const float* input = (const float*)d_in[0];
  const float* W_ih  = (const float*)d_in[1];
  const float* W_hh  = (const float*)d_in[2];
  const float* b_ih  = (const float*)d_in[3];
  const float* b_hh  = (const float*)d_in[4];
  const float* W_lin = (const float*)d_in[5];
  const float* b_lin = (const float*)d_in[6];
  const int T = in_sizes[0] / 3;

  hipLaunchKernelGGL(lstm_scan_kernel, dim3(1), dim3(64), 0, stream,
                     input, W_ih, W_hh, b_ih, b_hh, W_lin, b_lin,
                     (float*)d_out, T);
}